// LivingLooper_5815385718826
// MI455X (gfx1250) — compile-verified
//
#include <hip/hip_runtime.h>
#include <hip/hip_bf16.h>
#include <math.h>

// ---------------- problem constants ----------------
#define NMEM   1000
#define NLOOPS 4
#define NLAT   64
#define NCTX   3
#define NFEAT  768            // NLOOPS*NCTX*NLAT
#define KP     1008           // max n_win (1000-3=997) padded to mult of 16

typedef float v2f __attribute__((ext_vector_type(2)));
typedef float v8f __attribute__((ext_vector_type(8)));

// header layout in ws (ints):
// [0]=do_fit [1]=i1 [2]=rec [3]=new_rec [4]=li [5]=loop_len [6]=n_win
__global__ void ll_header(const int* pi, const int* prec, const int* pli,
                          const int* pll, int* hdr) {
    if (threadIdx.x == 0 && blockIdx.x == 0) {
        int i1  = pi[0] - 1;
        int rec = prec[0];
        int li  = pli[0];
        int ll  = pll[0];
        hdr[0] = (i1 != li) && (li >= 0) && (ll > NCTX);
        hdr[1] = i1;
        hdr[2] = rec;
        hdr[3] = (rec + 1) % NMEM;
        hdr[4] = li;
        hdr[5] = ll;
        hdr[6] = ll - NCTX;
    }
}

// c = feats.mean(0)  (768)  and  b = y.mean(0)  (64)
__global__ void ll_center_bias(const float* __restrict__ mem,
                               float* __restrict__ cvec,
                               float* __restrict__ bvec,
                               const int* __restrict__ hdr) {
    if (!hdr[0]) return;
    int idx = blockIdx.x * blockDim.x + threadIdx.x;
    int n_win = hdr[6];
    int begin = hdr[2] - hdr[5] + 1;
    int li    = hdr[4];
    if (idx < NFEAT) {
        int f = idx;
        int l = f / (NLAT * NCTX);
        int r = f % (NLAT * NCTX);
        int k = r / NCTX;
        int cx = r % NCTX;
        int row = (begin + cx) % NMEM; if (row < 0) row += NMEM;
        float s = 0.f;
        for (int n = 0; n < n_win; ++n) {
            s += mem[row * (NLOOPS * NLAT) + l * NLAT + k];
            row++; if (row >= NMEM) row = 0;
        }
        cvec[f] = s / (float)n_win;
    } else if (idx < NFEAT + NLAT) {
        int k = idx - NFEAT;
        int row = (begin + NCTX) % NMEM; if (row < 0) row += NMEM;
        float s = 0.f;
        for (int n = 0; n < n_win; ++n) {
            float t = mem[row * (NLOOPS * NLAT) + li * NLAT + k];
            s += t * fabsf(t);          // |t|^2 * sign(t)
            row++; if (row >= NMEM) row = 0;
        }
        bvec[k] = s / (float)n_win;
    }
}

// X[n,f] = tanh((feat - c)/2), zero-padded rows n >= n_win  (KP x 768 row-major)
__global__ void ll_build_x(const float* __restrict__ mem,
                           const float* __restrict__ cvec,
                           float* __restrict__ X,
                           const int* __restrict__ hdr) {
    if (!hdr[0]) return;
    int idx = blockIdx.x * blockDim.x + threadIdx.x;   // KP*768 threads
    int n = idx / NFEAT, f = idx % NFEAT;
    float v = 0.f;
    if (n < hdr[6]) {
        int begin = hdr[2] - hdr[5] + 1;
        int l = f / (NLAT * NCTX);
        int r = f % (NLAT * NCTX);
        int k = r / NCTX;
        int cx = r % NCTX;
        int row = (begin + n + cx) % NMEM; if (row < 0) row += NMEM;
        v = tanhf((mem[row * (NLOOPS * NLAT) + l * NLAT + k] - cvec[f]) * 0.5f);
    }
    X[idx] = v;
}

// Yp[n,k] = y - b, zero-padded  (KP x 64 row-major)
__global__ void ll_build_y(const float* __restrict__ mem,
                           const float* __restrict__ bvec,
                           float* __restrict__ Yp,
                           const int* __restrict__ hdr) {
    if (!hdr[0]) return;
    int idx = blockIdx.x * blockDim.x + threadIdx.x;   // KP*64 threads
    int n = idx / NLAT, k = idx % NLAT;
    float v = 0.f;
    if (n < hdr[6]) {
        int begin = hdr[2] - hdr[5] + 1;
        int li = hdr[4];
        int row = (begin + n + NCTX) % NMEM; if (row < 0) row += NMEM;
        float t = mem[row * (NLOOPS * NLAT) + li * NLAT + k];
        v = t * fabsf(t) - bvec[k];
    }
    Yp[idx] = v;
}

// G = X^T X (768x768) via f32 WMMA 16x16x4; one wave per 16x16 tile.
// A = X^T (row tile tr), B = X (col tile tc), K swept over KP padded windows.
__global__ void ll_gemm_xtx(const float* __restrict__ X,
                            float* __restrict__ G,
                            const int* __restrict__ hdr) {
    if (!hdr[0]) return;
    int wave = (blockIdx.x * blockDim.x + threadIdx.x) >> 5;
    if (wave >= 48 * 48) return;
    int lane = threadIdx.x & 31;
    int tr = wave / 48, tc = wave % 48;
    int half = lane >> 4;        // lanes 16-31 carry K pair +2
    int l16  = lane & 15;
    const float* __restrict__ colA = X + tr * 16 + l16;  // lanes contiguous in f
    const float* __restrict__ colB = X + tc * 16 + l16;
    v8f acc = {};
    for (int k0 = 0; k0 < KP; k0 += 4) {
        int ka = k0 + 2 * half;
        v2f a, b;
        a.x = colA[(size_t)(ka    ) * NFEAT];
        a.y = colA[(size_t)(ka + 1) * NFEAT];
        b.x = colB[(size_t)(ka    ) * NFEAT];
        b.y = colB[(size_t)(ka + 1) * NFEAT];
        acc = __builtin_amdgcn_wmma_f32_16x16x4_f32(
                  false, a, false, b, (short)0, acc, false, false);
    }
#pragma unroll
    for (int v = 0; v < 8; ++v) {
        int m = v + 8 * half;                 // C/D layout: VGPR v -> row v (+8)
        float val = acc[v];
        if (tr == tc && m == l16) val += 1e-4f;  // tiny ridge for SPD Cholesky
        G[(size_t)(tr * 16 + m) * NFEAT + tc * 16 + l16] = val;
    }
}

// R = X^T Yp (768x64) via f32 WMMA 16x16x4
__global__ void ll_gemm_xty(const float* __restrict__ X,
                            const float* __restrict__ Yp,
                            float* __restrict__ R,
                            const int* __restrict__ hdr) {
    if (!hdr[0]) return;
    int wave = (blockIdx.x * blockDim.x + threadIdx.x) >> 5;
    if (wave >= 48 * 4) return;
    int lane = threadIdx.x & 31;
    int tr = wave / 4, tc = wave % 4;
    int half = lane >> 4;
    int l16  = lane & 15;
    const float* __restrict__ colA = X  + tr * 16 + l16;
    const float* __restrict__ colB = Yp + tc * 16 + l16;
    v8f acc = {};
    for (int k0 = 0; k0 < KP; k0 += 4) {
        int ka = k0 + 2 * half;
        v2f a, b;
        a.x = colA[(size_t)(ka    ) * NFEAT];
        a.y = colA[(size_t)(ka + 1) * NFEAT];
        b.x = colB[(size_t)(ka    ) * NLAT];
        b.y = colB[(size_t)(ka + 1) * NLAT];
        acc = __builtin_amdgcn_wmma_f32_16x16x4_f32(
                  false, a, false, b, (short)0, acc, false, false);
    }
#pragma unroll
    for (int v = 0; v < 8; ++v) {
        int m = v + 8 * half;
        R[(size_t)(tr * 16 + m) * NLAT + tc * 16 + l16] = acc[v];
    }
}

// in-place right-looking Cholesky of G (lower triangle), one workgroup
__global__ void ll_cholesky(float* __restrict__ G, const int* __restrict__ hdr) {
    if (!hdr[0]) return;
    const int N = NFEAT;
    __shared__ float sdiag;
    int lane = threadIdx.x & 31;
    int wid  = threadIdx.x >> 5;
    int nw   = blockDim.x >> 5;
    for (int j = 0; j < N; ++j) {
        if (threadIdx.x == 0) {
            float d = sqrtf(fmaxf(G[(size_t)j * N + j], 1e-20f));
            G[(size_t)j * N + j] = d;
            sdiag = d;
        }
        __syncthreads();
        float inv = 1.0f / sdiag;
        for (int r = j + 1 + threadIdx.x; r < N; r += blockDim.x)
            G[(size_t)r * N + j] *= inv;
        __syncthreads();
        for (int r = j + 1 + wid; r < N; r += nw) {
            float lrj = G[(size_t)r * N + j];
            for (int c = j + 1 + lane; c <= r; c += 32)
                G[(size_t)r * N + c] = fmaf(-lrj, G[(size_t)c * N + j],
                                            G[(size_t)r * N + c]);
        }
        __syncthreads();
    }
}

// solve L L^T W = R ; 64 independent RHS columns, one thread each
__global__ void ll_trisolve(const float* __restrict__ G,
                            const float* __restrict__ R,
                            float* __restrict__ W,
                            const int* __restrict__ hdr) {
    if (!hdr[0]) return;
    int k = threadIdx.x;
    if (k >= NLAT) return;
    const int N = NFEAT;
    for (int j = 0; j < N; ++j) {               // forward: L y = R
        float s = R[(size_t)j * NLAT + k];
        for (int p = 0; p < j; ++p)
            s = fmaf(-G[(size_t)j * N + p], W[(size_t)p * NLAT + k], s);
        W[(size_t)j * NLAT + k] = s / G[(size_t)j * N + j];
    }
    for (int j = N - 1; j >= 0; --j) {          // backward: L^T w = y
        float s = W[(size_t)j * NLAT + k];
        for (int p = j + 1; p < N; ++p)
            s = fmaf(-G[(size_t)p * N + j], W[(size_t)p * NLAT + k], s);
        W[(size_t)j * NLAT + k] = s / G[(size_t)j * N + j];
    }
}

// scatter fitted row into the output weights/center/bias/mask
__global__ void ll_commit(const float* __restrict__ W,
                          const float* __restrict__ cvec,
                          const float* __restrict__ bvec,
                          float* __restrict__ oW, float* __restrict__ oC,
                          float* __restrict__ oB, float* __restrict__ oM,
                          const int* __restrict__ hdr) {
    if (!hdr[0]) return;
    int li = hdr[4];
    int idx = blockIdx.x * blockDim.x + threadIdx.x;
    const int NW = NFEAT * NLAT;
    if (idx < NW) {
        oW[(size_t)li * NW + idx] = W[idx];
    } else if (idx < NW + NFEAT) {
        oC[li * NFEAT + (idx - NW)] = cvec[idx - NW];
    } else if (idx < NW + NFEAT + NLAT) {
        oB[li * NLAT + (idx - NW - NFEAT)] = bvec[idx - NW - NFEAT];
    } else if (idx == NW + NFEAT + NLAT) {
        oM[li] = 1.0f;
    }
}

// evaluate all loops against the 3-frame feature; write out + new memory row
__global__ void ll_eval(const float* __restrict__ mem,
                        const float* __restrict__ z,
                        const float* __restrict__ oW,
                        const float* __restrict__ oC,
                        const float* __restrict__ oB,
                        const float* __restrict__ oM,
                        float* __restrict__ out0,
                        float* __restrict__ outMem,
                        const int* __restrict__ hdr) {
    __shared__ float sf[NFEAT];
    int i1 = hdr[1], rec = hdr[2], nrec = hdr[3];
    for (int f = threadIdx.x; f < NFEAT; f += blockDim.x) {
        int l = f / (NLAT * NCTX);
        int r = f % (NLAT * NCTX);
        int k = r / NCTX;
        int cx = r % NCTX;
        int row = (rec - (NCTX - 1) + cx) % NMEM; if (row < 0) row += NMEM;
        sf[f] = mem[row * (NLOOPS * NLAT) + l * NLAT + k];
    }
    __syncthreads();
    int tid = threadIdx.x;                 // 256 == NLOOPS*NLAT
    int j = tid >> 6, k = tid & 63;
    float zval;
    if (j == i1) {
        zval = z[i1 * NLAT + k];
    } else {
        float y = oB[j * NLAT + k];
        const float* cj = oC + j * NFEAT;
        const float* wj = oW + (size_t)j * NFEAT * NLAT + k;
        for (int f = 0; f < NFEAT; ++f) {
            float xv = tanhf((sf[f] - cj[f]) * 0.5f);
            y = fmaf(xv, wj[(size_t)f * NLAT], y);
        }
        zval = copysignf(sqrtf(fabsf(y)), y);     // |y|^0.5 * sign(y)
    }
    out0[tid] = zval * oM[j];
    outMem[nrec * (NLOOPS * NLAT) + tid] = zval;
}

extern "C" void kernel_launch(void* const* d_in, const int* in_sizes, int n_in,
                              void* d_out, int out_size, void* d_ws, size_t ws_size,
                              hipStream_t stream) {
    (void)in_sizes; (void)n_in; (void)out_size; (void)ws_size;
    const int*   d_i   = (const int*)d_in[0];
    const float* d_mem = (const float*)d_in[1];
    const float* d_z   = (const float*)d_in[2];
    const float* d_wt  = (const float*)d_in[3];
    const float* d_ct  = (const float*)d_in[4];
    const float* d_bs  = (const float*)d_in[5];
    const float* d_mk  = (const float*)d_in[6];
    const int*   d_rec = (const int*)d_in[7];
    const int*   d_li  = (const int*)d_in[8];
    const int*   d_ll  = (const int*)d_in[9];

    // output layout (floats): out(256) | memory(256000) | weights(196608) |
    //                         center(3072) | bias(256) | mask(4)
    float* out   = (float*)d_out;
    float* o_out = out;
    float* o_mem = out + 256;
    float* o_w   = o_mem + NMEM * NLOOPS * NLAT;
    float* o_c   = o_w + NLOOPS * NFEAT * NLAT;
    float* o_b   = o_c + NLOOPS * NFEAT;
    float* o_m   = o_b + NLOOPS * NLAT;

    // workspace layout (bytes)
    char*  ws   = (char*)d_ws;
    int*   hdr  = (int*)ws;                          // 64 B
    float* cvec = (float*)(ws + 256);                // 768 f
    float* bvec = (float*)(ws + 3328);               // 64 f
    float* X    = (float*)(ws + 4096);               // KP*768 f
    float* Yp   = (float*)(ws + 4096 + (size_t)KP * NFEAT * 4);
    float* G    = (float*)((char*)Yp + (size_t)KP * NLAT * 4);
    float* R    = (float*)((char*)G + (size_t)NFEAT * NFEAT * 4);
    float* W    = (float*)((char*)R + (size_t)NFEAT * NLAT * 4);
    // total ~6.1 MB

    // pass-through copies (fit/eval kernels overwrite the changed pieces)
    hipMemcpyAsync(o_mem, d_mem, (size_t)NMEM * NLOOPS * NLAT * 4,
                   hipMemcpyDeviceToDevice, stream);
    hipMemcpyAsync(o_w, d_wt, (size_t)NLOOPS * NFEAT * NLAT * 4,
                   hipMemcpyDeviceToDevice, stream);
    hipMemcpyAsync(o_c, d_ct, (size_t)NLOOPS * NFEAT * 4,
                   hipMemcpyDeviceToDevice, stream);
    hipMemcpyAsync(o_b, d_bs, (size_t)NLOOPS * NLAT * 4,
                   hipMemcpyDeviceToDevice, stream);
    hipMemcpyAsync(o_m, d_mk, (size_t)NLOOPS * 4,
                   hipMemcpyDeviceToDevice, stream);

    ll_header<<<1, 1, 0, stream>>>(d_i, d_rec, d_li, d_ll, hdr);
    ll_center_bias<<<4, 256, 0, stream>>>(d_mem, cvec, bvec, hdr);
    ll_build_x<<<(KP * NFEAT) / 256, 256, 0, stream>>>(d_mem, cvec, X, hdr);
    ll_build_y<<<(KP * NLAT) / 256, 256, 0, stream>>>(d_mem, bvec, Yp, hdr);
    ll_gemm_xtx<<<(48 * 48) / 8, 256, 0, stream>>>(X, G, hdr);
    ll_gemm_xty<<<(48 * 4) / 8, 256, 0, stream>>>(X, Yp, R, hdr);
    ll_cholesky<<<1, 256, 0, stream>>>(G, hdr);
    ll_trisolve<<<1, 64, 0, stream>>>(G, R, W, hdr);
    ll_commit<<<(NFEAT * NLAT + NFEAT + NLAT + 1 + 255) / 256, 256, 0, stream>>>(
        W, cvec, bvec, o_w, o_c, o_b, o_m, hdr);
    ll_eval<<<1, 256, 0, stream>>>(d_mem, d_z, o_w, o_c, o_b, o_m,
                                   o_out, o_mem, hdr);
}